// GCN_69114613729151
// MI455X (gfx1250) — compile-verified
//
#include <hip/hip_runtime.h>
#include <cmath>

typedef __attribute__((ext_vector_type(2))) float v2f;
typedef __attribute__((ext_vector_type(8))) float v8f;

#define NFEAT 512
#define NHID  16
#define NCLS  40

// ---------------------------------------------------------------------------
// GEMM1: sup1[N,16] = X[N,512] @ W1[512,16]  (f32 WMMA 16x16x4, K-loop of 128)
// One wave32 per 16-row tile. A-frag: lanes 0-15 -> M=lane, VGPR{0,1} -> K,K+1
// (lanes 16-31 get K+2,K+3). B-frag mirrored with N on lanes.
// X is streamed once (205 MB > 192 MB L2) -> non-temporal loads to avoid
// evicting the reused working set (W, sup, agg, edge arrays).
// ---------------------------------------------------------------------------
__global__ __launch_bounds__(256)
void gcn_gemm1_wmma(const float* __restrict__ X, const float* __restrict__ W1,
                    float* __restrict__ sup1, int nrows)
{
    const int lane  = threadIdx.x & 31;
    const int wave  = blockIdx.x * (blockDim.x >> 5) + (threadIdx.x >> 5);
    const int ntile = (nrows + 15) >> 4;
    if (wave >= ntile) return;                 // wave-uniform: EXEC stays all-1s

    const int row0  = wave << 4;
    const int r     = lane & 15;               // M (for A) / N (for B)
    const int khalf = (lane >> 4) << 1;        // 0 for lanes 0-15, 2 for 16-31
    int arow = row0 + r; if (arow >= nrows) arow = nrows - 1;
    const v2f* __restrict__ xr2 =
        (const v2f*)(X + (size_t)arow * NFEAT + khalf);

    v8f acc = {};
#pragma unroll 8
    for (int k0 = 0; k0 < NFEAT; k0 += 4) {
        const int k = k0 + khalf;
        v2f a = __builtin_nontemporal_load(xr2 + (k0 >> 1)); // A[r][k], A[r][k+1]
        v2f b;
        b.x = W1[(size_t)k * NHID + r];        // B[k][r]
        b.y = W1[(size_t)(k + 1) * NHID + r];  // B[k+1][r]
        acc = __builtin_amdgcn_wmma_f32_16x16x4_f32(false, a, false, b,
                                                    (short)0, acc, false, false);
    }

    // D layout: VGPR v -> M=v (lanes 0-15) / M=v+8 (lanes 16-31), N=lane&15
    const int mbase = (lane >> 4) << 3;
    float* __restrict__ dp = sup1 + (size_t)(row0 + mbase) * NHID + r;
    if (row0 + 16 <= nrows) {                  // wave-uniform fast path
#pragma unroll
        for (int v = 0; v < 8; ++v) dp[(size_t)v * NHID] = acc[v];
    } else {
#pragma unroll
        for (int v = 0; v < 8; ++v)
            if (row0 + mbase + v < nrows) dp[(size_t)v * NHID] = acc[v];
    }
}

// ---------------------------------------------------------------------------
// GEMM2: sup2[N,40] = relu(hpre[N,16]) @ W2[16,40]   (relu fused into A-frag)
// 3 N-tiles of 16 (cols 40..47 zero-padded). One wave per (rowtile, ntile).
// ---------------------------------------------------------------------------
__global__ __launch_bounds__(256)
void gcn_gemm2_wmma(const float* __restrict__ hpre, const float* __restrict__ W2,
                    float* __restrict__ sup2, int nrows)
{
    const int lane   = threadIdx.x & 31;
    const int wave   = blockIdx.x * (blockDim.x >> 5) + (threadIdx.x >> 5);
    const int rtiles = (nrows + 15) >> 4;
    if (wave >= rtiles * 3) return;            // wave-uniform

    const int ntile = wave % 3;
    const int rtile = wave / 3;
    const int row0  = rtile << 4;
    const int r     = lane & 15;
    const int n     = ntile * 16 + r;          // output column, valid if < 40
    const int khalf = (lane >> 4) << 1;
    int arow = row0 + r; if (arow >= nrows) arow = nrows - 1;
    const float* __restrict__ hr = hpre + (size_t)arow * NHID;

    v8f acc = {};
#pragma unroll
    for (int k0 = 0; k0 < NHID; k0 += 4) {
        const int k = k0 + khalf;
        v2f a, b;
        a.x = fmaxf(hr[k],     0.0f);          // relu fused (VALU select only)
        a.y = fmaxf(hr[k + 1], 0.0f);
        b.x = (n < NCLS) ? W2[(size_t)k * NCLS + n]       : 0.0f;
        b.y = (n < NCLS) ? W2[(size_t)(k + 1) * NCLS + n] : 0.0f;
        acc = __builtin_amdgcn_wmma_f32_16x16x4_f32(false, a, false, b,
                                                    (short)0, acc, false, false);
    }

    if (n < NCLS) {
        const int mbase = (lane >> 4) << 3;
        float* __restrict__ dp = sup2 + (size_t)(row0 + mbase) * NCLS + n;
        if (row0 + 16 <= nrows) {
#pragma unroll
            for (int v = 0; v < 8; ++v) dp[(size_t)v * NCLS] = acc[v];
        } else {
#pragma unroll
            for (int v = 0; v < 8; ++v)
                if (row0 + mbase + v < nrows) dp[(size_t)v * NCLS] = acc[v];
        }
    }
}

// ---------------------------------------------------------------------------
// Seed aggregation buffer with bias: agg[i*C + c] = b[c]
// ---------------------------------------------------------------------------
template <int C>
__global__ __launch_bounds__(256)
void gcn_init_bias(float* __restrict__ agg, const float* __restrict__ b, int n)
{
    const long long t = (long long)blockIdx.x * blockDim.x + threadIdx.x;
    if (t < (long long)n * C) agg[t] = b[(int)(t % C)];
}

// ---------------------------------------------------------------------------
// Edge scatter-add: agg[dst] += w_e * sup[src]; one thread per (edge, float4)
// ---------------------------------------------------------------------------
template <int C>
__global__ __launch_bounds__(256)
void gcn_edge_agg(const int* __restrict__ src, const int* __restrict__ dst,
                  const float* __restrict__ w, const float* __restrict__ sup,
                  float* __restrict__ agg, int nedges)
{
    constexpr int G = C / 4;                   // float4 groups per edge
    const long long t = (long long)blockIdx.x * blockDim.x + threadIdx.x;
    const long long e = t / G;
    const int       g = (int)(t - e * G);
    if (e >= nedges) return;

    const int   s  = src[e];
    const int   d  = dst[e];
    const float we = w[e];
    const float4 val = ((const float4*)(sup + (size_t)s * C))[g];
    float* __restrict__ ap = agg + (size_t)d * C + (size_t)g * 4;
    __hip_atomic_fetch_add(ap + 0, we * val.x, __ATOMIC_RELAXED, __HIP_MEMORY_SCOPE_AGENT);
    __hip_atomic_fetch_add(ap + 1, we * val.y, __ATOMIC_RELAXED, __HIP_MEMORY_SCOPE_AGENT);
    __hip_atomic_fetch_add(ap + 2, we * val.z, __ATOMIC_RELAXED, __HIP_MEMORY_SCOPE_AGENT);
    __hip_atomic_fetch_add(ap + 3, we * val.w, __ATOMIC_RELAXED, __HIP_MEMORY_SCOPE_AGENT);
}

// ---------------------------------------------------------------------------
// In-place per-row log-softmax over 40 classes (one thread per row)
// ---------------------------------------------------------------------------
__global__ __launch_bounds__(256)
void gcn_logsoftmax(float* __restrict__ out, int nrows)
{
    const int r = blockIdx.x * blockDim.x + threadIdx.x;
    if (r >= nrows) return;
    float* __restrict__ row = out + (size_t)r * NCLS;

    float v[NCLS];
    float m = -INFINITY;
#pragma unroll
    for (int c = 0; c < NCLS; ++c) { v[c] = row[c]; m = fmaxf(m, v[c]); }
    float s = 0.0f;
#pragma unroll
    for (int c = 0; c < NCLS; ++c) s += expf(v[c] - m);
    const float lse = m + logf(s);
#pragma unroll
    for (int c = 0; c < NCLS; ++c) row[c] = v[c] - lse;
}

// ---------------------------------------------------------------------------
extern "C" void kernel_launch(void* const* d_in, const int* in_sizes, int n_in,
                              void* d_out, int out_size, void* d_ws, size_t ws_size,
                              hipStream_t stream)
{
    const float* X    = (const float*)d_in[0];
    const int*   esrc = (const int*)  d_in[1];
    const int*   edst = (const int*)  d_in[2];
    const float* ew   = (const float*)d_in[3];
    const float* W1   = (const float*)d_in[4];
    const float* b1   = (const float*)d_in[5];
    const float* W2   = (const float*)d_in[6];
    const float* b2   = (const float*)d_in[7];
    float*       out  = (float*)d_out;

    const int N = in_sizes[0] / NFEAT;   // 100000
    const int E = in_sizes[1];           // 3200000

    // workspace layout (28.8 MB): sup1[N,16] | agg1[N,16] | sup2[N,40]
    float* sup1 = (float*)d_ws;
    float* agg1 = sup1 + (size_t)N * NHID;
    float* sup2 = agg1 + (size_t)N * NHID;
    // layer-2 aggregation accumulates directly into d_out

    // 1) sup1 = X @ W1
    {
        const int ntiles = (N + 15) / 16;
        const int blocks = (ntiles + 7) / 8;               // 8 waves / block
        gcn_gemm1_wmma<<<blocks, 256, 0, stream>>>(X, W1, sup1, N);
    }
    // 2) agg1 = b1 (so post-scatter agg1 = segment_sum + b1)
    {
        const long long tot = (long long)N * NHID;
        gcn_init_bias<NHID><<<(int)((tot + 255) / 256), 256, 0, stream>>>(agg1, b1, N);
    }
    // 3) agg1[dst] += w * sup1[src]
    {
        const long long tot = (long long)E * (NHID / 4);
        gcn_edge_agg<NHID><<<(int)((tot + 255) / 256), 256, 0, stream>>>(
            esrc, edst, ew, sup1, agg1, E);
    }
    // 4) sup2 = relu(agg1) @ W2
    {
        const int waves  = ((N + 15) / 16) * 3;
        const int blocks = (waves + 7) / 8;
        gcn_gemm2_wmma<<<blocks, 256, 0, stream>>>(agg1, W2, sup2, N);
    }
    // 5) out = b2
    {
        const long long tot = (long long)N * NCLS;
        gcn_init_bias<NCLS><<<(int)((tot + 255) / 256), 256, 0, stream>>>(out, b2, N);
    }
    // 6) out[dst] += w * sup2[src]
    {
        const long long tot = (long long)E * (NCLS / 4);
        gcn_edge_agg<NCLS><<<(int)((tot + 255) / 256), 256, 0, stream>>>(
            esrc, edst, ew, sup2, out, E);
    }
    // 7) in-place log_softmax
    gcn_logsoftmax<<<(N + 255) / 256, 256, 0, stream>>>(out, N);
}